// MultiHeadGENLayer_76596446757007
// MI455X (gfx1250) — compile-verified
//
#include <hip/hip_runtime.h>
#include <math.h>

#define N_NODES 100000
#define N_EDGES 600000
#define D       128
#define DH      256
#define EPS_MSG 1e-7f
#define EPS_BN  1e-5f
#define EPS_LN  1e-5f

typedef __attribute__((ext_vector_type(2))) float v2f;
typedef __attribute__((ext_vector_type(8))) float v8f;

// ---------- helpers ----------

__device__ __forceinline__ void atomicMaxF(float* addr, float val) {
    // sign-split trick: int max for >=0, uint min for <0 (init must be -inf).
    if (val >= 0.0f) {
        atomicMax((int*)addr, __float_as_int(val));
    } else {
        atomicMin((unsigned int*)addr, __float_as_uint(val));
    }
}

// msg = relu(x[src] + edge_attr @ W_edge + b_edge) + eps, 4 channels at c0
__device__ __forceinline__ void compute_msg(const float* __restrict__ x,
                                            float a0, float a1,
                                            const float* __restrict__ We,
                                            const float* __restrict__ be,
                                            int src, int c0, float m[4]) {
    float4 w0 = *(const float4*)(We + c0);
    float4 w1 = *(const float4*)(We + D + c0);
    float4 bb = *(const float4*)(be + c0);
    float4 xv = *(const float4*)(x + (size_t)src * D + c0);
    m[0] = fmaxf(xv.x + fmaf(a0, w0.x, fmaf(a1, w1.x, bb.x)), 0.0f) + EPS_MSG;
    m[1] = fmaxf(xv.y + fmaf(a0, w0.y, fmaf(a1, w1.y, bb.y)), 0.0f) + EPS_MSG;
    m[2] = fmaxf(xv.z + fmaf(a0, w0.z, fmaf(a1, w1.z, bb.z)), 0.0f) + EPS_MSG;
    m[3] = fmaxf(xv.w + fmaf(a0, w0.w, fmaf(a1, w1.w, bb.w)), 0.0f) + EPS_MSG;
}

// ---------- kernels ----------

__global__ void init_kernel(float* wmax, float* wden, float* wagg, float* sums) {
    size_t i = (size_t)blockIdx.x * blockDim.x + threadIdx.x;
    if (i < (size_t)N_NODES * D) {
        wmax[i] = __uint_as_float(0xFF800000u);  // -inf
        wden[i] = 0.0f;
        wagg[i] = 0.0f;
    }
    if (i < 2 * DH) sums[i] = 0.0f;
}

// One edge per wave (grid exact: 75000 blocks * 8 waves = 600000 edges).
// Edge-level values are forced into SGPRs via readfirstlane.

// pass 1: per-dst per-channel max of logits
__global__ void edge_max_kernel(const float* __restrict__ x,
                                const long long* __restrict__ ei,
                                const float* __restrict__ ea,
                                const float* __restrict__ We,
                                const float* __restrict__ be,
                                const float* __restrict__ tptr,
                                float* __restrict__ wmax) {
    int e = __builtin_amdgcn_readfirstlane(blockIdx.x * 8 + (threadIdx.x >> 5));
    int c0 = (threadIdx.x & 31) << 2;
    int src = __builtin_amdgcn_readfirstlane((int)ei[e]);
    int dst = __builtin_amdgcn_readfirstlane((int)ei[N_EDGES + e]);
    float a0 = ea[2 * e];
    float a1 = ea[2 * e + 1];
    float t = tptr[0];
    float m[4];
    compute_msg(x, a0, a1, We, be, src, c0, m);
    float* p = wmax + (size_t)dst * D + c0;
    atomicMaxF(p + 0, m[0] * t);
    atomicMaxF(p + 1, m[1] * t);
    atomicMaxF(p + 2, m[2] * t);
    atomicMaxF(p + 3, m[3] * t);
}

// pass 2: denom = sum exp(logit - max)
__global__ void edge_den_kernel(const float* __restrict__ x,
                                const long long* __restrict__ ei,
                                const float* __restrict__ ea,
                                const float* __restrict__ We,
                                const float* __restrict__ be,
                                const float* __restrict__ tptr,
                                const float* __restrict__ wmax,
                                float* __restrict__ wden) {
    int e = __builtin_amdgcn_readfirstlane(blockIdx.x * 8 + (threadIdx.x >> 5));
    int c0 = (threadIdx.x & 31) << 2;
    int src = __builtin_amdgcn_readfirstlane((int)ei[e]);
    int dst = __builtin_amdgcn_readfirstlane((int)ei[N_EDGES + e]);
    float a0 = ea[2 * e];
    float a1 = ea[2 * e + 1];
    float t = tptr[0];
    float m[4];
    compute_msg(x, a0, a1, We, be, src, c0, m);
    float4 mx = *(const float4*)(wmax + (size_t)dst * D + c0);
    float* p = wden + (size_t)dst * D + c0;
    atomicAdd(p + 0, __expf(m[0] * t - mx.x));
    atomicAdd(p + 1, __expf(m[1] * t - mx.y));
    atomicAdd(p + 2, __expf(m[2] * t - mx.z));
    atomicAdd(p + 3, __expf(m[3] * t - mx.w));
}

// pass 3: agg = sum msg * exp(logit - max) / denom
__global__ void edge_agg_kernel(const float* __restrict__ x,
                                const long long* __restrict__ ei,
                                const float* __restrict__ ea,
                                const float* __restrict__ We,
                                const float* __restrict__ be,
                                const float* __restrict__ tptr,
                                const float* __restrict__ wmax,
                                const float* __restrict__ wden,
                                float* __restrict__ wagg) {
    int e = __builtin_amdgcn_readfirstlane(blockIdx.x * 8 + (threadIdx.x >> 5));
    int c0 = (threadIdx.x & 31) << 2;
    int src = __builtin_amdgcn_readfirstlane((int)ei[e]);
    int dst = __builtin_amdgcn_readfirstlane((int)ei[N_EDGES + e]);
    float a0 = ea[2 * e];
    float a1 = ea[2 * e + 1];
    float t = tptr[0];
    float m[4];
    compute_msg(x, a0, a1, We, be, src, c0, m);
    float4 mx = *(const float4*)(wmax + (size_t)dst * D + c0);
    float4 dn = *(const float4*)(wden + (size_t)dst * D + c0);
    float* p = wagg + (size_t)dst * D + c0;
    atomicAdd(p + 0, m[0] * __expf(m[0] * t - mx.x) / dn.x);
    atomicAdd(p + 1, m[1] * __expf(m[1] * t - mx.y) / dn.y);
    atomicAdd(p + 2, m[2] * __expf(m[2] * t - mx.z) / dn.z);
    atomicAdd(p + 3, m[3] * __expf(m[3] * t - mx.w) / dn.w);
}

__global__ void residual_kernel(float* __restrict__ agg, const float* __restrict__ x) {
    size_t i = (size_t)blockIdx.x * blockDim.x + threadIdx.x;
    if (i < (size_t)N_NODES * D) agg[i] += x[i];
}

// C[M,NTN*16] = A[M,K] @ B[K,NTN*16] + bias via V_WMMA_F32_16X16X4_F32.
// Block = 8 waves covering one 16-row tile and 8 consecutive 16-col tiles.
// The shared 16xK A tile is staged in LDS once per block (8x reuse),
// padded by 4 floats/row so the 32-lane b64 fragment reads hit all 64 banks.
template <int LDA, int LDB, int K, int NTN>
__global__ void gemm_wmma_kernel(const float* __restrict__ A,
                                 const float* __restrict__ B,
                                 const float* __restrict__ bias,
                                 float* __restrict__ C) {
    constexpr int LDS_STRIDE = K + 4;
    __shared__ float sA[16 * LDS_STRIDE];

    int block_tile = blockIdx.x * 8;
    int mt = block_tile / NTN;
    int nt0 = block_tile - mt * NTN;  // multiple of 8
    int m0 = mt * 16;

    // cooperative A-tile load: 16*K floats, float4 per thread iteration
    constexpr int CHUNKS = 16 * (K / 4);  // 512 (K=128) or 1024 (K=256)
    for (int idx = threadIdx.x; idx < CHUNKS; idx += 256) {
        int r = idx / (K / 4);
        int cc = (idx - r * (K / 4)) * 4;
        float4 v = *(const float4*)(A + (size_t)(m0 + r) * LDA + cc);
        float* dstp = sA + r * LDS_STRIDE + cc;
        dstp[0] = v.x; dstp[1] = v.y; dstp[2] = v.z; dstp[3] = v.w;
    }
    __syncthreads();

    int wave = threadIdx.x >> 5;
    int lane = threadIdx.x & 31;
    int n0 = (nt0 + wave) * 16;
    int half = lane >> 4;   // 0: K pair k0..k0+1 ; 1: K pair k0+2..k0+3
    int lrow = lane & 15;

    const float* arow = sA + lrow * LDS_STRIDE;
    const float* bcol = B + n0 + lrow;

    v8f acc = {};
#pragma unroll 4
    for (int k0 = 0; k0 < K; k0 += 4) {
        int kk = k0 + 2 * half;
        // A fragment (16x4): lane<16 -> K=k0,k0+1 ; lane>=16 -> K=k0+2,k0+3
        v2f a = *(const v2f*)(arow + kk);
        // B fragment (4x16) row-striped: VGPR0 = rows k0/k0+2, VGPR1 = rows k0+1/k0+3
        v2f b;
        b.x = bcol[kk * LDB];
        b.y = bcol[(kk + 1) * LDB];
        acc = __builtin_amdgcn_wmma_f32_16x16x4_f32(
            false, a, false, b, (short)0, acc, false, false);
    }

    int col = n0 + lrow;
    float bv = bias[col];
#pragma unroll
    for (int r = 0; r < 8; ++r) {
        int row = m0 + r + half * 8;  // C layout: VGPR r -> M=r (lanes 0-15), M=r+8 (lanes 16-31)
        C[(size_t)row * (NTN * 16) + col] = acc[r] + bv;
    }
}

// per-column sum / sumsq over rows (coalesced: thread t owns column t)
__global__ void bn_stats_kernel(const float* __restrict__ h, float* __restrict__ sums) {
    int col = threadIdx.x;
    int r0 = blockIdx.x * 256;
    int rend = r0 + 256;
    if (rend > N_NODES) rend = N_NODES;
    float s = 0.0f, ss = 0.0f;
    for (int r = r0; r < rend; ++r) {
        float v = h[(size_t)r * DH + col];
        s += v;
        ss += v * v;
    }
    atomicAdd(&sums[col], s);
    atomicAdd(&sums[DH + col], ss);
}

__global__ void bn_finalize_kernel(const float* __restrict__ sums,
                                   const float* __restrict__ gamma,
                                   const float* __restrict__ beta,
                                   float* __restrict__ scale,
                                   float* __restrict__ shift) {
    int c = threadIdx.x;
    float mu = sums[c] * (1.0f / N_NODES);
    float var = sums[DH + c] * (1.0f / N_NODES) - mu * mu;
    float sc = gamma[c] * rsqrtf(var + EPS_BN);
    scale[c] = sc;
    shift[c] = beta[c] - mu * sc;
}

__global__ void bn_apply_kernel(float* __restrict__ h,
                                const float* __restrict__ scale,
                                const float* __restrict__ shift) {
    size_t i = (size_t)blockIdx.x * blockDim.x + threadIdx.x;
    if (i >= (size_t)N_NODES * DH) return;
    int c = (int)(i & (DH - 1));
    h[i] = fmaxf(fmaf(h[i], scale[c], shift[c]), 0.0f);
}

// LayerNorm over D=128 per row + ELU; one row per 128-thread block
__global__ void ln_elu_kernel(const float* __restrict__ g,
                              const float* __restrict__ gamma,
                              const float* __restrict__ beta,
                              float* __restrict__ out) {
    __shared__ float s1[D];
    __shared__ float s2[D];
    int row = blockIdx.x;
    int t = threadIdx.x;
    float v = g[(size_t)row * D + t];
    s1[t] = v;
    s2[t] = v * v;
    __syncthreads();
    for (int off = 64; off > 0; off >>= 1) {
        if (t < off) {
            s1[t] += s1[t + off];
            s2[t] += s2[t + off];
        }
        __syncthreads();
    }
    float mu = s1[0] * (1.0f / D);
    float var = s2[0] * (1.0f / D) - mu * mu;
    float y = (v - mu) * rsqrtf(var + EPS_LN) * gamma[t] + beta[t];
    out[(size_t)row * D + t] = y > 0.0f ? y : expm1f(y);
}

// ---------- launch ----------

extern "C" void kernel_launch(void* const* d_in, const int* in_sizes, int n_in,
                              void* d_out, int out_size, void* d_ws, size_t ws_size,
                              hipStream_t stream) {
    const float*     x     = (const float*)d_in[0];
    const long long* ei    = (const long long*)d_in[1];   // int64 [2, E]
    const float*     ea    = (const float*)d_in[2];
    const float*     We    = (const float*)d_in[3];
    const float*     be    = (const float*)d_in[4];
    const float*     tptr  = (const float*)d_in[5];
    const float*     W1    = (const float*)d_in[6];
    const float*     b1    = (const float*)d_in[7];
    const float*     bn_g  = (const float*)d_in[8];
    const float*     bn_b  = (const float*)d_in[9];
    const float*     W2    = (const float*)d_in[10];
    const float*     b2    = (const float*)d_in[11];
    const float*     ln_g  = (const float*)d_in[12];
    const float*     ln_b  = (const float*)d_in[13];
    float*           out   = (float*)d_out;

    float* ws     = (float*)d_ws;
    float* wmax   = ws;                                 // N*D
    float* wden   = wmax + (size_t)N_NODES * D;         // N*D
    float* wagg   = wden + (size_t)N_NODES * D;         // N*D
    float* wh     = wagg + (size_t)N_NODES * D;         // N*DH
    float* wg     = wh + (size_t)N_NODES * DH;          // N*D
    float* wsums  = wg + (size_t)N_NODES * D;           // 2*DH (sum, sumsq)
    float* wscale = wsums + 2 * DH;                     // DH
    float* wshift = wscale + DH;                        // DH

    const int nd_blocks   = (N_NODES * D + 255) / 256;          // 50000
    const int edge_blocks = N_EDGES / 8;                        // 75000 (1 edge/wave)
    const int ndh_blocks  = (N_NODES * DH + 255) / 256;         // 100000

    init_kernel<<<nd_blocks, 256, 0, stream>>>(wmax, wden, wagg, wsums);

    edge_max_kernel<<<edge_blocks, 256, 0, stream>>>(x, ei, ea, We, be, tptr, wmax);
    edge_den_kernel<<<edge_blocks, 256, 0, stream>>>(x, ei, ea, We, be, tptr, wmax, wden);
    edge_agg_kernel<<<edge_blocks, 256, 0, stream>>>(x, ei, ea, We, be, tptr, wmax, wden, wagg);

    residual_kernel<<<nd_blocks, 256, 0, stream>>>(wagg, x);

    // GEMM1: [N,128] @ [128,256] -> wh   (tiles: 6250 x 16, 8 waves/block)
    gemm_wmma_kernel<D, DH, D, DH / 16>
        <<<(N_NODES / 16) * (DH / 16) / 8, 256, 0, stream>>>(wagg, W1, b1, wh);

    bn_stats_kernel<<<(N_NODES + 255) / 256, 256, 0, stream>>>(wh, wsums);
    bn_finalize_kernel<<<1, DH, 0, stream>>>(wsums, bn_g, bn_b, wscale, wshift);
    bn_apply_kernel<<<ndh_blocks, 256, 0, stream>>>(wh, wscale, wshift);

    // GEMM2: [N,256] @ [256,128] -> wg   (tiles: 6250 x 8, 8 waves/block)
    gemm_wmma_kernel<DH, D, DH, D / 16>
        <<<(N_NODES / 16) * (D / 16) / 8, 256, 0, stream>>>(wh, W2, b2, wg);

    ln_elu_kernel<<<N_NODES, D, 0, stream>>>(wg, ln_g, ln_b, out);
}